// AGCRNCell_53128745451568
// MI455X (gfx1250) — compile-verified
//
#include <hip/hip_runtime.h>
#include <hip/hip_bf16.h>
#include <math.h>

// ---------------------------------------------------------------------------
// AGCRN cell for MI455X (gfx1250): fp32 WMMA (v_wmma_f32_16x16x4_f32) GEMMs,
// with double-buffered GLOBAL_LOAD_ASYNC_TO_LDS_B128 tile staging (ASYNCcnt).
// B=64, N=2048, DIN=2, H=64, cheb K=3, embed D=16.
// ---------------------------------------------------------------------------

typedef __attribute__((ext_vector_type(2))) float v2f;
typedef __attribute__((ext_vector_type(8))) float v8f;
typedef __attribute__((ext_vector_type(4))) int v4i;

__device__ __forceinline__ v8f wmma_f32(v2f a, v2f b, v8f c) {
  // 8 args: (neg_a, A, neg_b, B, c_mod, C, reuse_a, reuse_b)
  return __builtin_amdgcn_wmma_f32_16x16x4_f32(false, a, false, b, (short)0, c,
                                               false, false);
}

#define NN 2048
#define BB 64
#define CC 66        // DIN + H
#define LDX 4224     // BB*CC
#define KI3 198      // 3*CC

#if defined(__AMDGCN__)
#define HAS_ASYNC_LDS __has_builtin(__builtin_amdgcn_global_load_async_to_lds_b128)
#else
#define HAS_ASYNC_LDS 0
#endif

#if HAS_ASYNC_LDS
typedef __attribute__((address_space(1))) v4i* as1_v4i;
typedef __attribute__((address_space(3))) v4i* as3_v4i;
#endif

// copy 16B global -> LDS (async on gfx1250; sync float4 fallback)
__device__ __forceinline__ void lds_cp16(float* dst, const float* src) {
#if HAS_ASYNC_LDS
  v4i* sg = (v4i*)(const_cast<float*>(src));
  v4i* dl = (v4i*)dst;
  __builtin_amdgcn_global_load_async_to_lds_b128((as1_v4i)sg, (as3_v4i)dl, 0, 0);
#else
  *reinterpret_cast<float4*>(dst) = *reinterpret_cast<const float4*>(src);
#endif
}

__device__ __forceinline__ void wait_async() {
#if defined(__AMDGCN__) && __has_builtin(__builtin_amdgcn_s_wait_asynccnt)
  __builtin_amdgcn_s_wait_asynccnt(0);
#elif HAS_ASYNC_LDS
  asm volatile("s_wait_asynccnt 0" ::: "memory");
#endif
}

// ---------------------------------------------------------------------------
// Kernel 1: supports = softmax(relu(E @ E^T)) * mask  (optional 2nd softmax)
// ---------------------------------------------------------------------------
__global__ __launch_bounds__(256) void build_supports_kernel(
    const float* __restrict__ E, const float* __restrict__ mask,
    const int* __restrict__ normalize, float* __restrict__ S) {
  const int n = blockIdx.x;
  const int tid = threadIdx.x;
  __shared__ float e[16];
  __shared__ float sred[256];
  if (tid < 16) e[tid] = E[n * 16 + tid];
  __syncthreads();

  float v[8];
#pragma unroll
  for (int j = 0; j < 8; ++j) {
    const int m = tid + j * 256;
    const float* em = &E[m * 16];
    float acc = 0.f;
#pragma unroll
    for (int d = 0; d < 16; ++d) acc += e[d] * em[d];
    v[j] = fmaxf(acc, 0.f);
  }

  float lmax = v[0];
#pragma unroll
  for (int j = 1; j < 8; ++j) lmax = fmaxf(lmax, v[j]);
  sred[tid] = lmax;
  __syncthreads();
  for (int s = 128; s > 0; s >>= 1) {
    if (tid < s) sred[tid] = fmaxf(sred[tid], sred[tid + s]);
    __syncthreads();
  }
  const float rmax = sred[0];
  __syncthreads();
  float lsum = 0.f;
#pragma unroll
  for (int j = 0; j < 8; ++j) {
    v[j] = __expf(v[j] - rmax);
    lsum += v[j];
  }
  sred[tid] = lsum;
  __syncthreads();
  for (int s = 128; s > 0; s >>= 1) {
    if (tid < s) sred[tid] += sred[tid + s];
    __syncthreads();
  }
  const float inv = 1.f / sred[0];
  __syncthreads();
#pragma unroll
  for (int j = 0; j < 8; ++j) {
    const int m = tid + j * 256;
    v[j] = v[j] * inv * mask[(size_t)n * NN + m];
  }

  if (*normalize != 0) {
    float lm2 = v[0];
#pragma unroll
    for (int j = 1; j < 8; ++j) lm2 = fmaxf(lm2, v[j]);
    sred[tid] = lm2;
    __syncthreads();
    for (int s = 128; s > 0; s >>= 1) {
      if (tid < s) sred[tid] = fmaxf(sred[tid], sred[tid + s]);
      __syncthreads();
    }
    const float rm2 = sred[0];
    __syncthreads();
    float ls2 = 0.f;
#pragma unroll
    for (int j = 0; j < 8; ++j) {
      v[j] = __expf(v[j] - rm2);
      ls2 += v[j];
    }
    sred[tid] = ls2;
    __syncthreads();
    for (int s = 128; s > 0; s >>= 1) {
      if (tid < s) sred[tid] += sred[tid + s];
      __syncthreads();
    }
    const float i2 = 1.f / sred[0];
    __syncthreads();
#pragma unroll
    for (int j = 0; j < 8; ++j) v[j] *= i2;
  }

#pragma unroll
  for (int j = 0; j < 8; ++j) {
    const int m = tid + j * 256;
    S[(size_t)n * NN + m] = v[j];
  }
}

// ---------------------------------------------------------------------------
// Kernel 2: fp32 WMMA GEMM  C[M x Ncols] = A[M x K] * B[K x Ncols]
// 128x128 block tile, K-tile 16, double-buffered async global->LDS staging.
// 8 waves, each owns 32x64 (2x4 subtiles of 16x16).
// mode==1: Chebyshev epilogue C = 2*C - I.
// LDS pads: A rows 20 floats (80B, 16B-aligned, conflict-free), B rows 136
// floats (544B, 16B-aligned; half-wave K+2 offset lands +16 banks).
// ---------------------------------------------------------------------------
__global__ __launch_bounds__(256) void gemm_f32_wmma(
    const float* __restrict__ A, const float* __restrict__ B,
    float* __restrict__ C, int K, int lda, int ldb, int ldc, int mode) {
  __shared__ float As[2][128][20];
  __shared__ float Bs[2][16][136];

  const int tid = threadIdx.x;
  const int lane = tid & 31;
  const int wid = tid >> 5;     // 0..7
  const int wm = wid & 3;       // 4 row-waves * 32 rows
  const int wn = wid >> 2;      // 2 col-waves * 64 cols
  const int row0 = blockIdx.y * 128;
  const int col0 = blockIdx.x * 128;
  const int lm = lane & 15;
  const int kb = (lane >> 4) * 2;  // lanes 16-31 carry K+2,K+3

  v8f acc[2][4] = {};

  auto stage = [&](int bi, int k0) {
    // A tile 128x16 : 512 x 16B chunks, 2 per thread
#pragma unroll
    for (int it = 0; it < 2; ++it) {
      const int idx = tid + it * 256;
      const int r = idx >> 2;
      const int c4 = (idx & 3) << 2;
      lds_cp16(&As[bi][r][c4], &A[(size_t)(row0 + r) * lda + k0 + c4]);
    }
    // B tile 16x128 : 512 x 16B chunks, 2 per thread
#pragma unroll
    for (int it = 0; it < 2; ++it) {
      const int idx = tid + it * 256;
      const int r = idx >> 5;
      const int c4 = (idx & 31) << 2;
      lds_cp16(&Bs[bi][r][c4], &B[(size_t)(k0 + r) * ldb + col0 + c4]);
    }
  };

  // prologue: stage tile 0
  stage(0, 0);
  wait_async();
  __syncthreads();

  int bi = 0;
  for (int k0 = 0; k0 < K; k0 += 16) {
    if (k0 + 16 < K) stage(bi ^ 1, k0 + 16);  // async copies overlap compute

#pragma unroll
    for (int kk = 0; kk < 16; kk += 4) {
      v2f af[2], bf[4];
#pragma unroll
      for (int i = 0; i < 2; ++i) {
        const int r = wm * 32 + i * 16 + lm;
        af[i].x = As[bi][r][kk + kb];
        af[i].y = As[bi][r][kk + kb + 1];
      }
#pragma unroll
      for (int j = 0; j < 4; ++j) {
        const int c = wn * 64 + j * 16 + lm;
        bf[j].x = Bs[bi][kk + kb][c];
        bf[j].y = Bs[bi][kk + kb + 1][c];
      }
#pragma unroll
      for (int i = 0; i < 2; ++i)
#pragma unroll
        for (int j = 0; j < 4; ++j) acc[i][j] = wmma_f32(af[i], bf[j], acc[i][j]);
    }

    wait_async();     // next tile's async copies (this wave) have landed
    __syncthreads();  // everyone done reading bi / staging bi^1
    bi ^= 1;
  }

  const int cr = (lane >> 4) * 8;  // C layout: lanes 16-31 hold M=8..15
  const int cc = lane & 15;
#pragma unroll
  for (int i = 0; i < 2; ++i)
#pragma unroll
    for (int j = 0; j < 4; ++j)
#pragma unroll
      for (int r = 0; r < 8; ++r) {
        const int gr = row0 + wm * 32 + i * 16 + cr + r;
        const int gc = col0 + wn * 64 + j * 16 + cc;
        float val = acc[i][j][r];
        if (mode == 1) val = 2.f * val - (gr == gc ? 1.f : 0.f);
        C[(size_t)gr * ldc + gc] = val;
      }
}

// ---------------------------------------------------------------------------
// Kernel 3: Xt[n, b*66 + c] = concat(x, state)[b, n, c]
// ---------------------------------------------------------------------------
__global__ __launch_bounds__(256) void build_xt_kernel(
    const float* __restrict__ x, const float* __restrict__ state,
    float* __restrict__ Xt) {
  const int idx = blockIdx.x * 256 + threadIdx.x;
  if (idx >= NN * BB * CC) return;
  const int n = idx / (BB * CC);
  const int r = idx % (BB * CC);
  const int b = r / CC;
  const int c = r % CC;
  const float v = (c < 2) ? x[((size_t)b * NN + n) * 2 + c]
                          : state[((size_t)b * NN + n) * 64 + (c - 2)];
  Xt[(size_t)n * LDX + r] = v;
}

// ---------------------------------------------------------------------------
// Kernel 4 (gate): per node n -> W_n = E[n]·gate_W, C = A(64x198)·W_n(198x128)
// + bias, sigmoid.  z*state streamed into the candidate X buffer, r stored.
// ---------------------------------------------------------------------------
__global__ __launch_bounds__(256) void node_mix_gate_kernel(
    const float* __restrict__ E, const float* __restrict__ Wp,
    const float* __restrict__ bp, const float* __restrict__ Xt,
    const float* __restrict__ Y1, const float* __restrict__ Y2,
    const float* __restrict__ x, const float* __restrict__ state,
    float* __restrict__ R, float* __restrict__ Xc) {
  const int n = blockIdx.x;
  const int tid = threadIdx.x;
  const int lane = tid & 31;
  const int wid = tid >> 5;
  const int wm = wid & 1;   // 2 row-waves * 32 rows (batch)
  const int wn = wid >> 1;  // 4 col-waves * 32 cols (output)
  const int lm = lane & 15;
  const int kb = (lane >> 4) * 2;
  const int DSTR = 3 * CC * 128;  // d-stride in gate_W

  __shared__ float e[16];
  __shared__ float Wn[16][136];
  __shared__ float As[64][20];

  if (tid < 16) e[tid] = E[n * 16 + tid];
  if (tid < 128) {  // candidate X: raw x columns
    const int b = tid >> 1, c = tid & 1;
    Xc[(size_t)n * LDX + b * CC + c] = x[((size_t)b * NN + n) * 2 + c];
  }
  __syncthreads();

  v8f acc[2][2] = {};

  for (int c0 = 0; c0 < 208; c0 += 16) {  // K padded 198 -> 208
    // W_n chunk: 16x128
#pragma unroll
    for (int it = 0; it < 8; ++it) {
      const int idx = tid + it * 256;
      const int kc = idx >> 7;
      const int o = idx & 127;
      const int ki = c0 + kc;
      float w = 0.f;
      if (ki < KI3) {
        const float* wp = &Wp[(size_t)ki * 128 + o];
#pragma unroll
        for (int d = 0; d < 16; ++d) w += e[d] * wp[(size_t)d * DSTR];
      }
      Wn[kc][o] = w;
    }
    // A chunk: 64x16 gathered from [Xt | Y1 | Y2]
#pragma unroll
    for (int it = 0; it < 4; ++it) {
      const int idx = tid + it * 256;
      const int b = idx >> 4;
      const int kc = idx & 15;
      const int ki = c0 + kc;
      float a = 0.f;
      if (ki < KI3) {
        const size_t base = (size_t)n * LDX + b * CC;
        if (ki < CC) a = Xt[base + ki];
        else if (ki < 2 * CC) a = Y1[base + ki - CC];
        else a = Y2[base + ki - 2 * CC];
      }
      As[b][kc] = a;
    }
    __syncthreads();
#pragma unroll
    for (int kk = 0; kk < 16; kk += 4) {
      v2f af[2], bf[2];
#pragma unroll
      for (int i = 0; i < 2; ++i) {
        const int r = wm * 32 + i * 16 + lm;
        af[i].x = As[r][kk + kb];
        af[i].y = As[r][kk + kb + 1];
      }
#pragma unroll
      for (int j = 0; j < 2; ++j) {
        const int c = wn * 32 + j * 16 + lm;
        bf[j].x = Wn[kk + kb][c];
        bf[j].y = Wn[kk + kb + 1][c];
      }
#pragma unroll
      for (int i = 0; i < 2; ++i)
#pragma unroll
        for (int j = 0; j < 2; ++j) acc[i][j] = wmma_f32(af[i], bf[j], acc[i][j]);
    }
    __syncthreads();
  }

  const int cr = (lane >> 4) * 8;
  const int cc = lane & 15;
#pragma unroll
  for (int i = 0; i < 2; ++i)
#pragma unroll
    for (int j = 0; j < 2; ++j) {
      const int o = wn * 32 + j * 16 + cc;
      float bias = 0.f;
#pragma unroll
      for (int d = 0; d < 16; ++d) bias += e[d] * bp[d * 128 + o];
#pragma unroll
      for (int r = 0; r < 8; ++r) {
        const int b = wm * 32 + i * 16 + cr + r;
        const float val = acc[i][j][r] + bias;
        const float sg = 1.f / (1.f + __expf(-val));
        if (o < 64) {  // z: stream z*state into candidate input
          const float st = state[((size_t)b * NN + n) * 64 + o];
          Xc[(size_t)n * LDX + b * CC + 2 + o] = sg * st;
        } else {  // r gate
          R[((size_t)b * NN + n) * 64 + (o - 64)] = sg;
        }
      }
    }
}

// ---------------------------------------------------------------------------
// Kernel 5 (update): hc = tanh(A·W_n + b); out = r*state + (1-r)*hc
// ---------------------------------------------------------------------------
__global__ __launch_bounds__(256) void node_mix_upd_kernel(
    const float* __restrict__ E, const float* __restrict__ Wp,
    const float* __restrict__ bp, const float* __restrict__ Xc,
    const float* __restrict__ Y1, const float* __restrict__ Y2,
    const float* __restrict__ R, const float* __restrict__ state,
    float* __restrict__ out) {
  const int n = blockIdx.x;
  const int tid = threadIdx.x;
  const int lane = tid & 31;
  const int wid = tid >> 5;
  const int wm = wid & 3;   // 4 row-waves * 16 rows
  const int wn = wid >> 2;  // 2 col-waves * 32 cols
  const int lm = lane & 15;
  const int kb = (lane >> 4) * 2;
  const int DSTR = 3 * CC * 64;

  __shared__ float e[16];
  __shared__ float Wn[16][72];
  __shared__ float As[64][20];

  if (tid < 16) e[tid] = E[n * 16 + tid];
  __syncthreads();

  v8f acc[2] = {};

  for (int c0 = 0; c0 < 208; c0 += 16) {
    // W_n chunk 16x64
#pragma unroll
    for (int it = 0; it < 4; ++it) {
      const int idx = tid + it * 256;
      const int kc = idx >> 6;
      const int o = idx & 63;
      const int ki = c0 + kc;
      float w = 0.f;
      if (ki < KI3) {
        const float* wp = &Wp[(size_t)ki * 64 + o];
#pragma unroll
        for (int d = 0; d < 16; ++d) w += e[d] * wp[(size_t)d * DSTR];
      }
      Wn[kc][o] = w;
    }
    // A chunk 64x16
#pragma unroll
    for (int it = 0; it < 4; ++it) {
      const int idx = tid + it * 256;
      const int b = idx >> 4;
      const int kc = idx & 15;
      const int ki = c0 + kc;
      float a = 0.f;
      if (ki < KI3) {
        const size_t base = (size_t)n * LDX + b * CC;
        if (ki < CC) a = Xc[base + ki];
        else if (ki < 2 * CC) a = Y1[base + ki - CC];
        else a = Y2[base + ki - 2 * CC];
      }
      As[b][kc] = a;
    }
    __syncthreads();
#pragma unroll
    for (int kk = 0; kk < 16; kk += 4) {
      v2f af, bf[2];
      const int r = wm * 16 + lm;
      af.x = As[r][kk + kb];
      af.y = As[r][kk + kb + 1];
#pragma unroll
      for (int j = 0; j < 2; ++j) {
        const int c = wn * 32 + j * 16 + lm;
        bf[j].x = Wn[kk + kb][c];
        bf[j].y = Wn[kk + kb + 1][c];
      }
#pragma unroll
      for (int j = 0; j < 2; ++j) acc[j] = wmma_f32(af, bf[j], acc[j]);
    }
    __syncthreads();
  }

  const int cr = (lane >> 4) * 8;
  const int cc = lane & 15;
#pragma unroll
  for (int j = 0; j < 2; ++j) {
    const int o = wn * 32 + j * 16 + cc;
    float bias = 0.f;
#pragma unroll
    for (int d = 0; d < 16; ++d) bias += e[d] * bp[d * 64 + o];
#pragma unroll
    for (int r = 0; r < 8; ++r) {
      const int b = wm * 16 + cr + r;
      const float hc = tanhf(acc[j][r] + bias);
      const size_t si = ((size_t)b * NN + n) * 64 + o;
      const float rr = R[si];
      out[si] = rr * state[si] + (1.f - rr) * hc;
    }
  }
}

// ---------------------------------------------------------------------------
extern "C" void kernel_launch(void* const* d_in, const int* in_sizes, int n_in,
                              void* d_out, int out_size, void* d_ws,
                              size_t ws_size, hipStream_t stream) {
  const float* x     = (const float*)d_in[0];  // (64,2048,2)
  const float* state = (const float*)d_in[1];  // (64,2048,64)
  const float* E     = (const float*)d_in[2];  // (2048,16)
  const float* mask  = (const float*)d_in[3];  // (2048,2048)
  const float* gW    = (const float*)d_in[4];  // (16,3,66,128)
  const float* gb    = (const float*)d_in[5];  // (16,128)
  const float* uW    = (const float*)d_in[6];  // (16,3,66,64)
  const float* ub    = (const float*)d_in[7];  // (16,64)
  const int*   norm  = (const int*)d_in[8];    // scalar
  float* out = (float*)d_out;

  float* ws = (float*)d_ws;
  float* S  = ws;
  float* S2 = S  + (size_t)NN * NN;
  float* Xg = S2 + (size_t)NN * NN;
  float* Xc = Xg + (size_t)NN * LDX;
  float* Y1 = Xc + (size_t)NN * LDX;
  float* Y2 = Y1 + (size_t)NN * LDX;
  float* R  = Y2 + (size_t)NN * LDX;

  // 1. supports
  build_supports_kernel<<<NN, 256, 0, stream>>>(E, mask, norm, S);
  // 2. S2 = 2*S@S - I   (2048x2048x2048)
  gemm_f32_wmma<<<dim3(16, 16), 256, 0, stream>>>(S, S, S2, NN, NN, NN, NN, 1);
  // 3. gate input transposed to (N, B*66)
  build_xt_kernel<<<(NN * BB * CC + 255) / 256, 256, 0, stream>>>(x, state, Xg);
  // 4. Y1 = S @ Xg, Y2 = S2 @ Xg   (2048x2048x4224 each)
  gemm_f32_wmma<<<dim3(33, 16), 256, 0, stream>>>(S,  Xg, Y1, NN, NN, LDX, LDX, 0);
  gemm_f32_wmma<<<dim3(33, 16), 256, 0, stream>>>(S2, Xg, Y2, NN, NN, LDX, LDX, 0);
  // 5. gate mix: z,r -> R and candidate input Xc (x | z*state)
  node_mix_gate_kernel<<<NN, 256, 0, stream>>>(E, gW, gb, Xg, Y1, Y2, x, state,
                                               R, Xc);
  // 6. candidate graph conv
  gemm_f32_wmma<<<dim3(33, 16), 256, 0, stream>>>(S,  Xc, Y1, NN, NN, LDX, LDX, 0);
  gemm_f32_wmma<<<dim3(33, 16), 256, 0, stream>>>(S2, Xc, Y2, NN, NN, LDX, LDX, 0);
  // 7. update mix + GRU combine -> d_out
  node_mix_upd_kernel<<<NN, 256, 0, stream>>>(E, uW, ub, Xc, Y1, Y2, R, state,
                                              out);
}